// SegmentMM_9852654977048
// MI455X (gfx1250) — compile-verified
//
#include <hip/hip_runtime.h>

// ---- CDNA5 (gfx1250) wave32 WMMA segment-matmul ----
// out[i] = A[i] @ B[inv[segA[i]]],  A:[N,64] f32, B:[S,64,64] f32, segA sorted.

typedef __attribute__((ext_vector_type(2))) float v2f;
typedef __attribute__((ext_vector_type(4))) float v4f;
typedef __attribute__((ext_vector_type(8))) float v8f;
typedef __attribute__((ext_vector_type(4))) unsigned int v4u;
typedef __attribute__((ext_vector_type(8))) int v8i;
typedef __attribute__((ext_vector_type(4))) int v4i;

#define KD 64                // K dimension
#define MD 64                // M (output cols)
#define ROWS_PER_BLOCK 128   // 8 waves x 16 rows
#define LDSA_STRIDE 68       // dwords per A row (64 + 4 pad): conflict-free ds_load_b64
#define LDSB_KP_STRIDE 160   // dwords per interleaved k-pair (128 + 32 pad):
                             // halves land on bank sets {0..31} vs {32..63}

#if __has_builtin(__builtin_amdgcn_tensor_load_to_lds) && \
    __has_builtin(__builtin_amdgcn_s_wait_tensorcnt)
#define USE_TDM 1
#else
#define USE_TDM 0
#endif

// Build inverse permutation of segment_id_B: inv[segB[j]] = j.
__global__ void seg_inv_kernel(const int* __restrict__ segB,
                               int* __restrict__ inv, int S) {
  int t = blockIdx.x * blockDim.x + threadIdx.x;
  if (t < S) inv[segB[t]] = t;
}

__global__ __launch_bounds__(256)
void segmm_wmma_kernel(const float* __restrict__ A,
                       const float* __restrict__ B,
                       const int* __restrict__ segA,
                       const int* __restrict__ inv,
                       float* __restrict__ out,
                       int N) {
  __shared__ float ldsA[ROWS_PER_BLOCK * LDSA_STRIDE]; // 34816 B
  __shared__ float ldsB[32 * LDSB_KP_STRIDE];          // 20480 B

  const int tid  = threadIdx.x;
  const int wave = tid >> 5;
  const int lane = tid & 31;
  const int h    = lane >> 4;   // half-wave select
  const int l15  = lane & 15;
  const int blockRow = blockIdx.x * ROWS_PER_BLOCK;

  // ================= Stage the 128x64 A tile into LDS =================
#if USE_TDM
  if (wave == 0) {
    // Tensor DMA descriptor (ISA 8.3-8.5): 2D f32 tensor, tile 64x128,
    // LDS padding 4 dwords every 64 dwords -> padded row stride 68 floats.
    unsigned long long ga =
        (unsigned long long)(const void*)(A + (size_t)blockRow * KD);
    // LDS aperture: low 32 bits of a flat LDS pointer are the LDS address.
    unsigned int ldsa = (unsigned int)(unsigned long long)(const void*)&ldsA[0];

    v4u g0;
    g0.x = 1u;                                      // count=1, user mode
    g0.y = ldsa;                                    // lds_addr
    g0.z = (unsigned int)(ga & 0xFFFFFFFFu);        // global_addr[31:0]
    g0.w = (unsigned int)((ga >> 32) & 0x1FFFFFFu)  // global_addr[56:32]
           | (2u << 30);                            // type = 2 ("image")

    v8i g1;
    g1[0] = (2 << 16)        // data_size = 4 bytes
          | (1 << 20)        // pad_enable
          | (5 << 22)        // pad_interval: 64 dwords
          | (3 << 25);       // pad_amount:   4 dwords
    g1[1] = (KD & 0xFFFF) << 16;                    // tensor_dim0 = 64 (lo16)
    g1[2] = (N & 0xFFFF) << 16;                     // dim0 hi=0 | tensor_dim1 lo16
    g1[3] = ((N >> 16) & 0xFFFF) | (KD << 16);      // tensor_dim1 hi16 | tile_dim0=64
    g1[4] = ROWS_PER_BLOCK;                         // tile_dim1 = 128, tile_dim2 = 0
    g1[5] = KD;                                     // tensor_dim0_stride = 64 elems
    g1[6] = 0;
    g1[7] = 0;

    v4i gz4 = {0, 0, 0, 0};
    v8i gz8 = {0, 0, 0, 0, 0, 0, 0, 0};
    // 6-arg form (clang-23 / therock-10.0 headers): groups 2/3 unused for 2D.
    __builtin_amdgcn_tensor_load_to_lds(g0, g1, gz4, gz4, gz8, 0);
    __builtin_amdgcn_s_wait_tensorcnt(0);
  }
#else
  {
    // Fallback: coalesced float4 loads, 8 per thread.
#pragma unroll
    for (int i = 0; i < 8; ++i) {
      int f   = tid + i * 256;
      int row = f >> 4;
      int c4  = f & 15;
      int grow = blockRow + row;
      if (grow >= N) grow = N - 1;
      v4f v = *(const v4f*)(A + (size_t)grow * KD + c4 * 4);
      *(v4f*)(&ldsA[row * LDSA_STRIDE + c4 * 4]) = v;
    }
  }
#endif
  __syncthreads();

  const int r0  = wave * 16;
  const int gr0 = blockRow + r0;

  // Segment ranges (segA sorted) as SGPRs: block range drives the uniform
  // loop/barriers, wave range gates the compute pass.
  int blast = blockRow + ROWS_PER_BLOCK - 1; if (blast >= N) blast = N - 1;
  int wfirst = gr0 < N ? gr0 : N - 1;
  int wlast  = gr0 + 15; if (wlast >= N) wlast = N - 1;
  const int bs_lo = __builtin_amdgcn_readfirstlane(segA[blockRow < N ? blockRow : N - 1]);
  const int bs_hi = __builtin_amdgcn_readfirstlane(segA[blast]);
  const int ws_lo = __builtin_amdgcn_readfirstlane(segA[wfirst]);
  const int ws_hi = __builtin_amdgcn_readfirstlane(segA[wlast]);
  const bool block_uniform = (bs_lo == bs_hi);

  // ---- Preload all 16 A fragments from LDS (ds_load_b64, conflict-free) ----
  // A-frag (16x4 f32): lanes 0-15 row=lane, V0=k0,V1=k1; lanes 16-31 V0=k2,V1=k3.
  v2f afrag[16];
#pragma unroll
  for (int kk = 0; kk < 16; ++kk) {
    const float* p = &ldsA[(r0 + l15) * LDSA_STRIDE + kk * 4 + h * 2];
    afrag[kk] = *(const v2f*)p;
  }

  v8f acc[4];
#pragma unroll
  for (int nt = 0; nt < 4; ++nt) acc[nt] = v8f{};

  for (int s = bs_lo; s <= bs_hi; ++s) {
    // ---- Stage B_s into LDS, pair-interleaved: all 256 threads ----
    // ldsB[p*160 + 2n] = { B[2p][n], B[2p+1][n] }  (p = k-pair, n = column)
    {
      int bIdx = __builtin_amdgcn_readfirstlane(inv[s]);
      const float* Bs = B + (size_t)bIdx * (KD * MD);
#pragma unroll
      for (int t = 0; t < 2; ++t) {
        int task = tid + t * 256;     // 0..511
        int p    = task >> 4;         // k-pair 0..31
        int n4   = task & 15;         // float4 column group
        v4f ra = *(const v4f*)(Bs + (size_t)(2 * p)     * MD + n4 * 4);
        v4f rb = *(const v4f*)(Bs + (size_t)(2 * p + 1) * MD + n4 * 4);
        float* d = &ldsB[p * LDSB_KP_STRIDE + n4 * 8];
        v4f w0 = {ra.x, rb.x, ra.y, rb.y};
        v4f w1 = {ra.z, rb.z, ra.w, rb.w};
        *(v4f*)(d)     = w0;
        *(v4f*)(d + 4) = w1;
      }
    }
    __syncthreads();

    // Wave-uniform skip if this stripe has no rows in segment s
    // (EXEC stays all-1s inside, as WMMA requires).
    if (s >= ws_lo && s <= ws_hi) {
      v8f P[4];
#pragma unroll
      for (int nt = 0; nt < 4; ++nt) P[nt] = v8f{};

      // K = 64 as 16 chained K=4 WMMAs, 4 independent column tiles for ILP.
#pragma unroll
      for (int kk = 0; kk < 16; ++kk) {
#pragma unroll
        for (int nt = 0; nt < 4; ++nt) {
          // One ds_load_b64 yields {B[k][n], B[k+1][n]}, k = kk*4 + h*2.
          const float* bp =
              &ldsB[(kk * 2 + h) * LDSB_KP_STRIDE + (nt * 16 + l15) * 2];
          v2f b = *(const v2f*)bp;
          P[nt] = __builtin_amdgcn_wmma_f32_16x16x4_f32(
              false, afrag[kk], false, b, (short)0, P[nt], false, false);
        }
      }

      if (block_uniform) {
#pragma unroll
        for (int nt = 0; nt < 4; ++nt) acc[nt] = P[nt];
      } else {
        // Boundary block (rare): keep only rows whose segment == s.
        // C/D layout: vgpr v holds row (v + 8*h) of the tile.
#pragma unroll
        for (int v = 0; v < 8; ++v) {
          int rowg = gr0 + v + h * 8; if (rowg >= N) rowg = N - 1;
          bool take = (segA[rowg] == s);
#pragma unroll
          for (int nt = 0; nt < 4; ++nt)
            acc[nt][v] = take ? P[nt][v] : acc[nt][v];
        }
      }
    }

    if (s != bs_hi) __syncthreads();  // protect ldsB before restaging
  }

  // ---- Store: vgpr v, half h -> row gr0 + v + 8h; lane l15 -> col nt*16+l15.
#pragma unroll
  for (int v = 0; v < 8; ++v) {
    int row = gr0 + v + h * 8;
    if (row < N) {
#pragma unroll
      for (int nt = 0; nt < 4; ++nt)
        out[(size_t)row * MD + nt * 16 + l15] = acc[nt][v];
    }
  }
}

extern "C" void kernel_launch(void* const* d_in, const int* in_sizes, int n_in,
                              void* d_out, int out_size, void* d_ws, size_t ws_size,
                              hipStream_t stream) {
  const float* A    = (const float*)d_in[0];   // [N, 64]
  const float* Bm   = (const float*)d_in[1];   // [S, 64, 64]
  const int*   segA = (const int*)d_in[2];     // [N] sorted
  const int*   segB = (const int*)d_in[3];     // [S] permutation
  float* out = (float*)d_out;

  const int N = in_sizes[2];
  const int S = in_sizes[3];

  int* inv = (int*)d_ws;                       // S ints of scratch

  seg_inv_kernel<<<(S + 127) / 128, 128, 0, stream>>>(segB, inv, S);

  int blocks = (N + ROWS_PER_BLOCK - 1) / ROWS_PER_BLOCK;
  segmm_wmma_kernel<<<blocks, 256, 0, stream>>>(A, Bm, segA, inv, out, N);
}